// RegularisedOptimalTransport_35373350650679
// MI455X (gfx1250) — compile-verified
//
#include <hip/hip_runtime.h>
#include <hip/hip_bf16.h>
#include <stdint.h>

// Sinkhorn / entropic OT on MI455X (gfx1250).
// b=16, m=n=1024, eps=0.1, 20 iters. Cost tensor = 64MB, fits in 192MB L2.
// Row pass: TDM DMAs 8x1024 fp32 tiles into LDS; per-row lse reduction
// finishes on the matrix pipe via V_WMMA_F32_16X16X4_F32 (B = ones).
// Raw v_exp_f32 / v_log_f32 (no subnormal guards): exponents are <= 0 and
// underflow-to-zero is the desired behavior inside a logsumexp sum.

#define B    16
#define M    1024
#define N    1024
#define EPS        0.1f
#define INV_EPS    10.0f
#define LOG2E      1.4426950408889634f
#define LN2        0.6931471805599453f
#define SCALE      (INV_EPS * LOG2E)   // t -> base-2 exponent
#define EPS_LN2    (EPS * LN2)         // base-2 lse -> eps * natural lse
#define ROWS_PER_BLK 8

typedef unsigned int uint32x4 __attribute__((ext_vector_type(4)));
typedef int          int32x8  __attribute__((ext_vector_type(8)));
typedef int          int32x4  __attribute__((ext_vector_type(4)));
typedef float        v2f      __attribute__((ext_vector_type(2)));
typedef float        v8f      __attribute__((ext_vector_type(8)));

#if __has_builtin(__builtin_amdgcn_exp2f)
#define EXP2(x) __builtin_amdgcn_exp2f(x)     // raw v_exp_f32
#else
#define EXP2(x) exp2f(x)
#endif
#if __has_builtin(__builtin_amdgcn_logf)
#define LOG2(x) __builtin_amdgcn_logf(x)      // raw v_log_f32 (= log2)
#else
#define LOG2(x) __log2f(x)
#endif

// ---------------------------------------------------------------- init
__global__ void ot_init(const float* __restrict__ mu, const float* __restrict__ nu,
                        float* u, float* v, float* lmu, float* lnu,
                        float* errsq, int* done) {
    int i = blockIdx.x * blockDim.x + threadIdx.x;
    if (i < B * M) {
        u[i] = 0.0f;
        v[i] = 0.0f;
        lmu[i] = __logf(mu[i] + 1e-8f);
        lnu[i] = __logf(nu[i] + 1e-8f);
    }
    if (i < B) errsq[i] = 0.0f;
    if (i == 0) *done = 0;
}

// ------------------------------------------------- row pass (reduce over j)
// One block = 8 rows of one batch. TDM pulls the 8x1024 fp32 tile into LDS.
__global__ __launch_bounds__(256) void ot_row(const float* __restrict__ cost,
                                              const float* __restrict__ v,
                                              const float* __restrict__ lmu,
                                              float* u, float* errsq,
                                              const int* __restrict__ done) {
    __shared__ __align__(16) float sv[N];
    __shared__ __align__(16) float tile[ROWS_PER_BLK * N];

    const int b    = blockIdx.x >> 7;        // 128 row-blocks per batch
    const int row0 = (blockIdx.x & 127) * ROWS_PER_BLK;

    // stage v[b,:] into LDS
    for (int j = threadIdx.x; j < N; j += 256) sv[j] = v[b * N + j];

    const int wave = threadIdx.x >> 5;
    const int lane = threadIdx.x & 31;

#if __has_builtin(__builtin_amdgcn_tensor_load_to_lds)
    if (wave == 0) {
        // ---- Tensor DMA descriptor (D#), 2D tile 1024 x 8, fp32 ----
        uint64_t ga  = (uint64_t)(uintptr_t)(cost + ((size_t)b * M + row0) * (size_t)N);
        uint32_t lds = (uint32_t)(uintptr_t)(&tile[0]);   // LDS offset = addr[31:0]

        uint32x4 g0;
        g0.x = 1u;                                    // count=1, user descriptor
        g0.y = lds;                                   // lds_addr
        g0.z = (uint32_t)ga;                          // global_addr[31:0]
        g0.w = (uint32_t)(ga >> 32) | (2u << 30);     // global_addr[56:32] | type=2

        int32x8 g1;
        g1[0] = (2 << 16);                 // workgroup_mask=0, data_size=2 (4B)
        g1[1] = (int)(1024u << 16);        // tensor_dim0[15:0]=1024 in [31:16]
        g1[2] = (int)(1024u << 16);        // tensor_dim0 hi=0 | tensor_dim1[15:0]=1024
        g1[3] = (int)(1024u << 16);        // tensor_dim1 hi=0 | tile_dim0=1024
        g1[4] = ROWS_PER_BLK;              // tile_dim1=8, tile_dim2=0
        g1[5] = 1024;                      // tensor_dim0_stride[31:0]
        g1[6] = 0;                         // stride hi | tensor_dim1_stride lo
        g1[7] = 0;

        int32x4 g2 = {0, 0, 0, 0};
        int32x4 g3 = {0, 0, 0, 0};
        int32x8 g4 = {0, 0, 0, 0, 0, 0, 0, 0};

        __builtin_amdgcn_tensor_load_to_lds(g0, g1, g2, g3, g4, 0);
        __builtin_amdgcn_s_wait_tensorcnt(0);
    }
#else
    // fallback: cooperative copy
    for (int j = threadIdx.x; j < ROWS_PER_BLK * N; j += 256)
        tile[j] = cost[((size_t)b * M + row0) * (size_t)N + j];
#endif
    __syncthreads();

    // each wave reduces one row: register-resident two-pass logsumexp (base 2)
    const float* r = &tile[wave * N];
    float y[32];
    float m2 = -3.0e38f;
    #pragma unroll
    for (int k = 0; k < 8; ++k) {
        const int j = k * 128 + lane * 4;
        float4 c4 = *(const float4*)&r[j];
        float4 v4 = *(const float4*)&sv[j];
        y[k * 4 + 0] = (v4.x - c4.x) * SCALE;
        y[k * 4 + 1] = (v4.y - c4.y) * SCALE;
        y[k * 4 + 2] = (v4.z - c4.z) * SCALE;
        y[k * 4 + 3] = (v4.w - c4.w) * SCALE;
        m2 = fmaxf(m2, fmaxf(fmaxf(y[k*4+0], y[k*4+1]), fmaxf(y[k*4+2], y[k*4+3])));
    }
    float s2 = 0.0f;
    #pragma unroll
    for (int k = 0; k < 32; ++k) s2 += EXP2(y[k] - m2);

    // ---- cross-lane combine ----
    // 1) max over the 32 lanes (shuffle ladder)
    float mmax = m2;
    #pragma unroll
    for (int off = 16; off > 0; off >>= 1)
        mmax = fmaxf(mmax, __shfl_xor(mmax, off, 32));

    // 2) rescale each lane's partial once, then sum all 32 lanes on the
    //    matrix pipe: A[i,{0,2}] = partial of lane i / lane i+16, B = ones
    //    => every column of D holds e_i + e_{i+16}.
    float el = s2 * EXP2(m2 - mmax);
    v2f a;  a.x = el;   a.y = 0.0f;
    v2f bb; bb.x = 1.0f; bb.y = 1.0f;
    v8f c = {};
    v8f d = __builtin_amdgcn_wmma_f32_16x16x4_f32(
        /*neg_a=*/false, a, /*neg_b=*/false, bb,
        /*c_mod=*/(short)0, c, /*reuse_a=*/false, /*reuse_b=*/false);
    float tot = ((d[0] + d[1]) + (d[2] + d[3])) + ((d[4] + d[5]) + (d[6] + d[7]));
    tot += __shfl_xor(tot, 16, 32);   // rows 0-7 half + rows 8-15 half

    if (lane == 0) {
        const int gi = b * M + row0 + wave;
        float u_new = EPS * lmu[gi] - EPS_LN2 * (mmax + LOG2(tot));
        float du = u_new - u[gi];
        atomicAdd(&errsq[b], du * du);
        if (*done == 0) u[gi] = u_new;
    }
}

// ------------------------------------------------- col pass (reduce over i)
// One thread per column; chunked online logsumexp, coalesced column reads.
__global__ __launch_bounds__(256) void ot_col(const float* __restrict__ cost,
                                              const float* __restrict__ u,
                                              const float* __restrict__ lnu,
                                              float* v,
                                              const int* __restrict__ done) {
    __shared__ __align__(16) float su[M];
    const int b = blockIdx.x >> 2;           // 4 column-blocks per batch
    const int j = (blockIdx.x & 3) * 256 + threadIdx.x;

    for (int i = threadIdx.x; i < M; i += 256) su[i] = u[b * M + i];
    __syncthreads();

    const float* c = cost + (size_t)b * M * N + j;
    float m2 = -3.0e38f, s2 = 0.0f;
    for (int i0 = 0; i0 < M; i0 += 8) {
        float y[8];
        float mc = -3.0e38f;
        #pragma unroll
        for (int k = 0; k < 8; ++k) {
            float cv = c[(size_t)(i0 + k) * N];
            y[k] = (su[i0 + k] - cv) * SCALE;
            mc = fmaxf(mc, y[k]);
        }
        float sc = 0.0f;
        #pragma unroll
        for (int k = 0; k < 8; ++k) sc += EXP2(y[k] - mc);
        float mn = fmaxf(m2, mc);
        s2 = s2 * EXP2(m2 - mn) + sc * EXP2(mc - mn);
        m2 = mn;
    }

    const int gj = b * N + j;
    float v_new = EPS * lnu[gj] - EPS_LN2 * (m2 + LOG2(s2));
    if (*done == 0) v[gj] = v_new;
}

// -------------------------------------------- convergence flag update
__global__ void ot_done(float* errsq, int* done) {
    int t = threadIdx.x;
    float e = (t < B) ? sqrtf(errsq[t]) : 0.0f;
    #pragma unroll
    for (int off = 16; off > 0; off >>= 1) e += __shfl_xor(e, off, 32);
    if (t == 0) {
        if (e * (1.0f / (float)B) < 1e-6f) *done = 1;
    }
    if (t < B) errsq[t] = 0.0f;
}

// -------------------------------------------- P = exp((-c + u + v)/eps)
__global__ __launch_bounds__(256) void ot_out(const float* __restrict__ cost,
                                              const float* __restrict__ u,
                                              const float* __restrict__ v,
                                              float* __restrict__ P) {
    const int b = blockIdx.x >> 10;
    const int i = blockIdx.x & 1023;
    const float ui = u[b * M + i];
    const int j = threadIdx.x * 4;
    const size_t off = ((size_t)(b * M + i)) * (size_t)N + j;
    float4 c4 = *(const float4*)(cost + off);
    float4 v4 = *(const float4*)(v + b * N + j);
    float4 o;
    o.x = EXP2((ui + v4.x - c4.x) * SCALE);
    o.y = EXP2((ui + v4.y - c4.y) * SCALE);
    o.z = EXP2((ui + v4.z - c4.z) * SCALE);
    o.w = EXP2((ui + v4.w - c4.w) * SCALE);
    *(float4*)(P + off) = o;
}

// ---------------------------------------------------------------- launch
extern "C" void kernel_launch(void* const* d_in, const int* in_sizes, int n_in,
                              void* d_out, int out_size, void* d_ws, size_t ws_size,
                              hipStream_t stream) {
    const float* cost = (const float*)d_in[0];
    const float* mu   = (const float*)d_in[1];
    const float* nu   = (const float*)d_in[2];
    float* out = (float*)d_out;

    float* ws    = (float*)d_ws;
    float* u     = ws;                 // 16*1024
    float* v     = ws + 16384;         // 16*1024
    float* lmu   = ws + 32768;         // 16*1024
    float* lnu   = ws + 49152;         // 16*1024
    float* errsq = ws + 65536;         // 16
    int*   done  = (int*)(ws + 65552); // 1

    ot_init<<<(B * M + 255) / 256, 256, 0, stream>>>(mu, nu, u, v, lmu, lnu, errsq, done);

    for (int it = 0; it < 20; ++it) {
        ot_row<<<B * (M / ROWS_PER_BLK), 256, 0, stream>>>(cost, v, lmu, u, errsq, done);
        ot_col<<<B * (N / 256), 256, 0, stream>>>(cost, u, lnu, v, done);
        ot_done<<<1, 32, 0, stream>>>(errsq, done);
    }

    ot_out<<<B * M, 256, 0, stream>>>(cost, u, v, out);
}